// Hybrid_CNN_LSTM_Residual_Inversion_60421599920184
// MI455X (gfx1250) — compile-verified
//
#include <hip/hip_runtime.h>
#include <hip/hip_bf16.h>

// ---------------------------------------------------------------------------
// Hybrid CNN + BiLSTM + LSTM-decoder for MI455X (gfx1250, wave32, WMMA).
// All GEMM work uses V_WMMA_F32_16X16X4_F32 (f32 in / f32 acc) so numerics
// match the f32 reference; the conv stage is HBM-bound (~0.7GB @ 23.3TB/s)
// and the scan is latency-bound, so the f32 matrix rate is not the limiter.
// ---------------------------------------------------------------------------

typedef __attribute__((ext_vector_type(2))) float v2f;
typedef __attribute__((ext_vector_type(8))) float v8f;

__device__ __forceinline__ v8f wmma_f32_k4(v2f a, v2f b, v8f c) {
  // (neg_a, A, neg_b, B, c_mod, C, reuse_a, reuse_b)
  return __builtin_amdgcn_wmma_f32_16x16x4_f32(false, a, false, b, (short)0, c,
                                               false, false);
}

__device__ __forceinline__ float sigf(float x) {
  return 1.f / (1.f + __expf(-x));
}

// ---------------------------------------------------------------------------
// conv1d (+bias, +relu) as implicit GEMM, Kw compile-time (3/5/7) so the
// (channel, tap) decode is branch-free constant-divisor arithmetic.
//   M = 16 time positions per wave (64 per block), N = 16 out channels,
//   K = Cin*Kw (always a multiple of 4 for this net).
// Input layouts: x_btc=1 -> x[b][t][c] (raw input), else x[b][c][t].
// Output written at channel offset out_coff inside a buffer with
// out_cstride total channels (implements the 3-way concat for free).
// ---------------------------------------------------------------------------
template <int KW>
__global__ __launch_bounds__(128) void conv1d_relu_wmma(
    const float* __restrict__ x, const float* __restrict__ w,
    const float* __restrict__ bias, float* __restrict__ y, int T, int Cin,
    int x_btc, int in_cstride, int out_coff, int out_cstride) {
  constexpr int W = 64 + KW - 1;  // staged window incl. halo
  constexpr int LDA = W + 1;      // odd leading dim -> conflict-free
  constexpr int pad = KW >> 1;
  __shared__ float xs[192 * LDA];  // up to Cin=192

  const int b = blockIdx.z;
  const int t0 = blockIdx.x * 64;
  const int nb = blockIdx.y * 16;

  // Stage the input window (with halo) into LDS.
  const int tot = Cin * W;
  for (int e = threadIdx.x; e < tot; e += 128) {
    int c = e / W, i = e - c * W;  // constant divisor
    int t = t0 - pad + i;
    float v = 0.f;
    if (t >= 0 && t < T)
      v = x_btc ? x[((size_t)b * T + t) * Cin + c]
                : x[((size_t)b * in_cstride + c) * T + t];
    xs[c * LDA + i] = v;
  }
  __syncthreads();

  const int wv = threadIdx.x >> 5;
  const int lane = threadIdx.x & 31;
  const int ln = lane & 15;  // A-row m / B-col n within tile
  const int lh = lane >> 4;  // K-pair select (A/B), M-half select (C/D)
  const int tl = wv * 16;    // this wave's local time base
  const int KC = Cin * KW;

  v8f acc = {0.f, 0.f, 0.f, 0.f, 0.f, 0.f, 0.f, 0.f};
  const float* wrow = w + (size_t)(nb + ln) * KC;  // w[n][c][q] == wrow[k]
  const float* arow = xs + tl + ln;
  const int kbase = 2 * lh;

#pragma unroll 4
  for (int kb = 0; kb < KC; kb += 4) {
    int k0 = kb + kbase;
    int c0 = k0 / KW;        // constant-divisor mul/shift, no divergence
    int c1 = (k0 + 1) / KW;
    v2f a, bf;
    // xs[c][ tl+ln+q ] with k = c*KW+q  ->  arow[k + c*(LDA-KW)]
    a.x = arow[k0 + c0 * (LDA - KW)];
    a.y = arow[k0 + 1 + c1 * (LDA - KW)];
    bf.x = wrow[k0];
    bf.y = wrow[k0 + 1];
    acc = wmma_f32_k4(a, bf, acc);
  }

  float bv = bias[nb + ln];
#pragma unroll
  for (int r = 0; r < 8; ++r) {
    int m = r + 8 * lh;
    float v = acc[r] + bv;
    v = v > 0.f ? v : 0.f;
    y[((size_t)b * out_cstride + out_coff + ln) * T + (t0 + tl + m)] = v;
  }
}

// ---------------------------------------------------------------------------
// Encoder input projection (hoisted out of the scan, fully parallel):
//   P[t][b][n] = sum_k h3[b][k][t] * wi[n][k] + bi[n] + bh[n]
// One workgroup per (t, direction); M=64 batch, N=128 gates, K=64.
// ---------------------------------------------------------------------------
__global__ __launch_bounds__(256) void lstm_xproj_wmma(
    const float* __restrict__ h3, const float* __restrict__ wif,
    const float* __restrict__ bif, const float* __restrict__ bhf,
    const float* __restrict__ wib, const float* __restrict__ bib,
    const float* __restrict__ bhb, float* __restrict__ Pf,
    float* __restrict__ Pb, int T) {
  __shared__ float xsh[64 * 65];
  const int t = blockIdx.x;
  const int dir = blockIdx.y;
  const float* W = dir ? wib : wif;
  const float* Bi = dir ? bib : bif;
  const float* Bh = dir ? bhb : bhf;
  float* P = dir ? Pb : Pf;

  for (int e = threadIdx.x; e < 64 * 64; e += 256) {
    int m = e >> 6, k = e & 63;
    xsh[m * 65 + k] = h3[((size_t)m * 64 + k) * T + t];
  }
  __syncthreads();

  const int wv = threadIdx.x >> 5, lane = threadIdx.x & 31;
  const int ln = lane & 15, lh = lane >> 4;
  float* Pt = P + (size_t)t * 64 * 128;

  for (int i = 0; i < 4; ++i) {
    int tile = wv * 4 + i;
    int tm = (tile >> 3) * 16, tn = (tile & 7) * 16;
    int gn = tn + ln;
    float bsum = Bi[gn] + Bh[gn];
    v8f acc = {bsum, bsum, bsum, bsum, bsum, bsum, bsum, bsum};
    const float* wrow = W + (size_t)gn * 64;
#pragma unroll 4
    for (int kb = 0; kb < 64; kb += 4) {
      int k0 = kb + 2 * lh;
      v2f a, bf;
      a.x = xsh[(tm + ln) * 65 + k0];
      a.y = xsh[(tm + ln) * 65 + k0 + 1];
      bf.x = wrow[k0];
      bf.y = wrow[k0 + 1];
      acc = wmma_f32_k4(a, bf, acc);
    }
#pragma unroll
    for (int r = 0; r < 8; ++r)
      Pt[(size_t)(tm + r + 8 * lh) * 128 + gn] = acc[r];
  }
}

// ---------------------------------------------------------------------------
// Sequential encoder scan. One workgroup (8 waves) per direction.
// Per step only h @ Wh^T remains (K=32 -> 8 WMMA/tile). Tile assignment:
// wave owns all 4 gate columns for its (tm, j-slice) so i/f/g/o land in the
// same lane -> gate fusion in registers, c-state register-resident.
// P[t] gate biases are double-buffered in registers: step s+1's 32 values
// per lane are loaded while step s computes, hiding HBM/L2 latency behind a
// full recurrence step.
// state layout: [hf | cf | hb | cb], each [64][32].
// ---------------------------------------------------------------------------
__global__ __launch_bounds__(256) void lstm_encoder_seq(
    const float* __restrict__ Pf, const float* __restrict__ Pb,
    const float* __restrict__ whf, const float* __restrict__ whb,
    float* __restrict__ state, int T) {
  __shared__ float hsh[64 * 34];   // h [m][k], lda 34
  __shared__ float wsh[32 * 128];  // Wh in B-layout [k][n]
  const int dir = blockIdx.x;
  const float* P = dir ? Pb : Pf;
  const float* wh = dir ? whb : whf;

  for (int e = threadIdx.x; e < 32 * 128; e += 256) {
    int k = e >> 7, n = e & 127;
    wsh[k * 128 + n] = wh[(size_t)n * 32 + k];
  }
  for (int e = threadIdx.x; e < 64 * 34; e += 256) hsh[e] = 0.f;
  __syncthreads();

  const int wv = threadIdx.x >> 5, lane = threadIdx.x & 31;
  const int ln = lane & 15, lh = lane >> 4;
  const int tm = (wv >> 1) * 16;     // batch-row tile
  const int j = (wv & 1) * 16 + ln;  // hidden index 0..31

  int rowoff[8];
#pragma unroll
  for (int r = 0; r < 8; ++r) rowoff[r] = (tm + r + 8 * lh) * 128 + j;

  float creg[8];
#pragma unroll
  for (int r = 0; r < 8; ++r) creg[r] = 0.f;

  // Preload P for step 0.
  float pnext[32];
  {
    const float* Pt = P + (size_t)(dir ? (T - 1) : 0) * 8192;
#pragma unroll
    for (int gi = 0; gi < 4; ++gi)
#pragma unroll
      for (int r = 0; r < 8; ++r) pnext[gi * 8 + r] = Pt[rowoff[r] + gi * 32];
  }

  for (int s = 0; s < T; ++s) {
    v8f g[4];
#pragma unroll
    for (int gi = 0; gi < 4; ++gi)
#pragma unroll
      for (int r = 0; r < 8; ++r) g[gi][r] = pnext[gi * 8 + r];

    // Issue next step's P loads now; they complete during this step.
    if (s + 1 < T) {
      const float* Ptn = P + (size_t)(dir ? (T - 2 - s) : (s + 1)) * 8192;
#pragma unroll
      for (int gi = 0; gi < 4; ++gi)
#pragma unroll
        for (int r = 0; r < 8; ++r)
          pnext[gi * 8 + r] = Ptn[rowoff[r] + gi * 32];
    }

#pragma unroll
    for (int gi = 0; gi < 4; ++gi) {
      int gn = gi * 32 + j;
#pragma unroll
      for (int kb = 0; kb < 32; kb += 4) {
        int k0 = kb + 2 * lh;
        v2f a, bf;
        a.x = hsh[(tm + ln) * 34 + k0];
        a.y = hsh[(tm + ln) * 34 + k0 + 1];
        bf.x = wsh[k0 * 128 + gn];
        bf.y = wsh[(k0 + 1) * 128 + gn];
        g[gi] = wmma_f32_k4(a, bf, g[gi]);
      }
    }
    __syncthreads();  // all reads of hsh done before overwrite
#pragma unroll
    for (int r = 0; r < 8; ++r) {
      float c = sigf(g[1][r]) * creg[r] + sigf(g[0][r]) * tanhf(g[2][r]);
      creg[r] = c;
      hsh[(tm + r + 8 * lh) * 34 + j] = sigf(g[3][r]) * tanhf(c);
    }
    __syncthreads();
  }

  float* hout = state + (size_t)dir * 2 * 2048;
  float* cout_ = hout + 2048;
#pragma unroll
  for (int r = 0; r < 8; ++r) {
    int m = tm + r + 8 * lh;
    hout[m * 32 + j] = hsh[m * 34 + j];
    cout_[m * 32 + j] = creg[r];
  }
}

// ---------------------------------------------------------------------------
// Decoder: 512 steps of 2 stacked LSTM cells + FC, single workgroup.
// A-tile LDS layout: cols 0..3 = y, 4..35 = h0, 36..67 = h1, so
// layer0 reads cols [0,36) (K=36) and layer1 reads cols [4,68) (K=64).
// Combined [Wi;Wh] B-matrices staged once in LDS.
// ---------------------------------------------------------------------------
__global__ __launch_bounds__(256) void lstm_decoder(
    const float* __restrict__ state, const float* __restrict__ w0i,
    const float* __restrict__ w0h, const float* __restrict__ b0i,
    const float* __restrict__ b0h, const float* __restrict__ w1i,
    const float* __restrict__ w1h, const float* __restrict__ b1i,
    const float* __restrict__ b1h, const float* __restrict__ fcw,
    const float* __restrict__ fcb, float* __restrict__ out, int L) {
  __shared__ float A[64 * 68];
  __shared__ float B0[36 * 128];
  __shared__ float B1[64 * 128];
  __shared__ float bb0[128], bb1[128];
  __shared__ float fcs[4 * 32 + 4];
  const int tid = threadIdx.x;

  for (int e = tid; e < 36 * 128; e += 256) {
    int k = e >> 7, n = e & 127;
    B0[e] = (k < 4) ? w0i[(size_t)n * 4 + k] : w0h[(size_t)n * 32 + (k - 4)];
  }
  for (int e = tid; e < 64 * 128; e += 256) {
    int k = e >> 7, n = e & 127;
    B1[e] = (k < 32) ? w1i[(size_t)n * 32 + k] : w1h[(size_t)n * 32 + (k - 32)];
  }
  for (int e = tid; e < 128; e += 256) {
    bb0[e] = b0i[e] + b0h[e];
    bb1[e] = b1i[e] + b1h[e];
  }
  for (int e = tid; e < 132; e += 256) fcs[e] = (e < 128) ? fcw[e] : fcb[e - 128];
  for (int e = tid; e < 64 * 68; e += 256) A[e] = 0.f;
  __syncthreads();
  for (int e = tid; e < 2048; e += 256) {
    int m = e >> 5, jj = e & 31;
    A[m * 68 + 4 + jj] = state[e];          // h0 <- hf
    A[m * 68 + 36 + jj] = state[4096 + e];  // h1 <- hb
  }
  __syncthreads();

  const int wv = tid >> 5, lane = tid & 31;
  const int ln = lane & 15, lh = lane >> 4;
  const int tm = (wv >> 1) * 16;
  const int j = (wv & 1) * 16 + ln;

  float c0reg[8], c1reg[8];
#pragma unroll
  for (int r = 0; r < 8; ++r) {
    int m = tm + r + 8 * lh;
    c0reg[r] = state[2048 + m * 32 + j];  // c0 <- cf
    c1reg[r] = state[6144 + m * 32 + j];  // c1 <- cb
  }

  for (int l = 0; l < L; ++l) {
    // ---- layer 0: gates = [y|h0] @ B0 + bb0, K = 36 ----
    v8f g[4];
#pragma unroll
    for (int gi = 0; gi < 4; ++gi) {
      int gn = gi * 32 + j;
      float bs = bb0[gn];
#pragma unroll
      for (int r = 0; r < 8; ++r) g[gi][r] = bs;
#pragma unroll
      for (int kb = 0; kb < 36; kb += 4) {
        int k0 = kb + 2 * lh;
        v2f a, bf;
        a.x = A[(tm + ln) * 68 + k0];
        a.y = A[(tm + ln) * 68 + k0 + 1];
        bf.x = B0[k0 * 128 + gn];
        bf.y = B0[(k0 + 1) * 128 + gn];
        g[gi] = wmma_f32_k4(a, bf, g[gi]);
      }
    }
    __syncthreads();
#pragma unroll
    for (int r = 0; r < 8; ++r) {
      float c = sigf(g[1][r]) * c0reg[r] + sigf(g[0][r]) * tanhf(g[2][r]);
      c0reg[r] = c;
      A[(tm + r + 8 * lh) * 68 + 4 + j] = sigf(g[3][r]) * tanhf(c);
    }
    __syncthreads();
    // ---- layer 1: gates = [h0|h1] @ B1 + bb1, K = 64 ----
#pragma unroll
    for (int gi = 0; gi < 4; ++gi) {
      int gn = gi * 32 + j;
      float bs = bb1[gn];
#pragma unroll
      for (int r = 0; r < 8; ++r) g[gi][r] = bs;
#pragma unroll
      for (int kb = 0; kb < 64; kb += 4) {
        int k0 = kb + 2 * lh;
        v2f a, bf;
        a.x = A[(tm + ln) * 68 + 4 + k0];
        a.y = A[(tm + ln) * 68 + 4 + k0 + 1];
        bf.x = B1[k0 * 128 + gn];
        bf.y = B1[(k0 + 1) * 128 + gn];
        g[gi] = wmma_f32_k4(a, bf, g[gi]);
      }
    }
    __syncthreads();
#pragma unroll
    for (int r = 0; r < 8; ++r) {
      float c = sigf(g[1][r]) * c1reg[r] + sigf(g[0][r]) * tanhf(g[2][r]);
      c1reg[r] = c;
      A[(tm + r + 8 * lh) * 68 + 36 + j] = sigf(g[3][r]) * tanhf(c);
    }
    __syncthreads();
    // ---- FC: y = h1 @ fc_w^T + fc_b ; feed back + emit ----
    {
      int m = tid >> 2, o = tid & 3;
      float acc = fcs[128 + o];
#pragma unroll
      for (int jj = 0; jj < 32; ++jj)
        acc += A[m * 68 + 36 + jj] * fcs[o * 32 + jj];
      out[((size_t)m * L + l) * 4 + o] = acc;
      A[m * 68 + o] = acc;
    }
    __syncthreads();
  }
}

// ---------------------------------------------------------------------------
// Host launch
// ---------------------------------------------------------------------------
extern "C" void kernel_launch(void* const* d_in, const int* in_sizes, int n_in,
                              void* d_out, int out_size, void* d_ws,
                              size_t ws_size, hipStream_t stream) {
  (void)n_in;
  (void)ws_size;
  const float* x = (const float*)d_in[0];
  const int T = in_sizes[0] / (64 * 8);
  const int L = out_size / (64 * 4);

#define F(i) ((const float*)d_in[(i)])
  // Workspace layout (stream-ordered aliasing):
  //   bufA  : [64, 64ch, T]  (block outputs)            @ 0        (64 MiB)
  //   bufB  : [64, 192ch, T] (concat buffers)           @ 64 MiB   (192 MiB)
  //   Pf/Pb : [T, 64, 128]   (alias bufB after convs)   @ 64 MiB / 192 MiB
  //   state : [4][64][32]                                @ 320 MiB
  char* ws = (char*)d_ws;
  float* bufA = (float*)(ws);
  float* bufB = (float*)(ws + 67108864ull);
  float* Pf = (float*)(ws + 67108864ull);
  float* Pb = (float*)(ws + 67108864ull + 134217728ull);
  float* state = (float*)(ws + 67108864ull + 268435456ull);

  auto conv = [&](const float* in, const float* wt, const float* bs, float* out,
                  int Cin, int Cout, int Kw, int btc, int incs, int coff,
                  int cstr) {
    dim3 g(T / 64, Cout / 16, 64), b(128);
    if (Kw == 3)
      conv1d_relu_wmma<3><<<g, b, 0, stream>>>(in, wt, bs, out, T, Cin, btc,
                                               incs, coff, cstr);
    else if (Kw == 5)
      conv1d_relu_wmma<5><<<g, b, 0, stream>>>(in, wt, bs, out, T, Cin, btc,
                                               incs, coff, cstr);
    else
      conv1d_relu_wmma<7><<<g, b, 0, stream>>>(in, wt, bs, out, T, Cin, btc,
                                               incs, coff, cstr);
  };

  // ---- block 1 (Cin=8 -> C=32) ----
  conv(x, F(3), F(4), bufB, 8, 32, 3, 1, 8, 0, 96);
  conv(x, F(5), F(6), bufB, 8, 32, 5, 1, 8, 32, 96);
  conv(x, F(7), F(8), bufB, 8, 32, 7, 1, 8, 64, 96);
  conv(bufB, F(9), F(10), bufA, 96, 32, 3, 0, 96, 0, 32);
  // ---- block 2 (32 -> 64) ----
  conv(bufA, F(11), F(12), bufB, 32, 64, 3, 0, 32, 0, 192);
  conv(bufA, F(13), F(14), bufB, 32, 64, 5, 0, 32, 64, 192);
  conv(bufA, F(15), F(16), bufB, 32, 64, 7, 0, 32, 128, 192);
  conv(bufB, F(17), F(18), bufA, 192, 64, 3, 0, 192, 0, 64);
  // ---- block 3 (64 -> 64) ----
  conv(bufA, F(19), F(20), bufB, 64, 64, 3, 0, 64, 0, 192);
  conv(bufA, F(21), F(22), bufB, 64, 64, 5, 0, 64, 64, 192);
  conv(bufA, F(23), F(24), bufB, 64, 64, 7, 0, 64, 128, 192);
  conv(bufB, F(25), F(26), bufA, 192, 64, 3, 0, 192, 0, 64);

  // ---- encoder: hoisted input projection, then sequential scan ----
  lstm_xproj_wmma<<<dim3(T, 2), dim3(256), 0, stream>>>(
      bufA, F(27), F(29), F(30), F(31), F(33), F(34), Pf, Pb, T);
  lstm_encoder_seq<<<dim3(2), dim3(256), 0, stream>>>(Pf, Pb, F(28), F(32),
                                                      state, T);
  // ---- decoder ----
  lstm_decoder<<<dim3(1), dim3(256), 0, stream>>>(
      state, F(35), F(36), F(37), F(38), F(39), F(40), F(41), F(42), F(43),
      F(44), (float*)d_out, L);
#undef F
}